// NMSOp_9371618640643
// MI455X (gfx1250) — compile-verified
//
#include <hip/hip_runtime.h>
#include <stdint.h>

#define N_BOX 2048
#define MASK_WORDS (N_BOX / 32) // 64 u32 per row
#define ROWS_PER_BLK 32         // sorted rows handled per block in kernel 3

typedef __attribute__((ext_vector_type(4))) unsigned int v4u;
typedef __attribute__((ext_vector_type(8))) int v8i;
typedef __attribute__((ext_vector_type(4))) int v4i;

// ---------------------------------------------------------------------------
// Kernel 1: corners + areas (contiguous in workspace so one TDM tile covers both)
// ---------------------------------------------------------------------------
__global__ void prep_kernel(const float* __restrict__ boxes,
                            float* __restrict__ corners,
                            float* __restrict__ area) {
  int i = blockIdx.x * blockDim.x + threadIdx.x;
  if (i >= N_BOX) return;
  const float* b = boxes + i * 10;
  float x = b[0], y = b[1], w = b[3], l = b[4], rt = b[6];
  float hl = 0.5f * l, hw = 0.5f * w;
  float c = cosf(rt), s = sinf(rt);
  const float lx[4] = {1.f, -1.f, -1.f, 1.f};
  const float ly[4] = {1.f, 1.f, -1.f, -1.f};
#pragma unroll
  for (int k = 0; k < 4; k++) {
    float p0 = lx[k] * hl, p1 = ly[k] * hw;
    corners[i * 8 + 2 * k + 0] = p0 * c - p1 * s + x;
    corners[i * 8 + 2 * k + 1] = p0 * s + p1 * c + y;
  }
  area[i] = w * l;
}

// ---------------------------------------------------------------------------
// Kernel 2: stable descending argsort via rank counting.
// ---------------------------------------------------------------------------
__global__ void rank_kernel(const float* __restrict__ boxes,
                            int* __restrict__ order) {
  __shared__ float ssc[N_BOX];
  for (int k = threadIdx.x; k < N_BOX; k += blockDim.x)
    ssc[k] = boxes[k * 10 + 9];
  __syncthreads();
  int i = blockIdx.x * blockDim.x + threadIdx.x;
  if (i >= N_BOX) return;
  float si = ssc[i];
  int r = 0;
  for (int j = 0; j < N_BOX; j++) {
    float sj = ssc[j];
    r += (sj > si) || (sj == si && j < i);
  }
  order[r] = i; // permutation (strict total order)
}

// ---------------------------------------------------------------------------
// Kernel 3: suppression bitmask. One TDM DMA stages corners+areas (72KB) into
// LDS per block; Sutherland-Hodgman ping-pong vertex buffers live in LDS with
// [vertex][thread] layout (conflict-free b64 access); wave32 ballot packs bits.
// ---------------------------------------------------------------------------
__global__ void __launch_bounds__(256)
iou_mask_kernel(const float* __restrict__ corners, // == workspace base
                const int* __restrict__ order,
                const float* __restrict__ thr_p,
                unsigned* __restrict__ mask) {
  __shared__ float sc[N_BOX * 8 + N_BOX]; // 64KB corners + 8KB areas
  __shared__ float2 ply[2][8][256];       // 32KB polygon ping-pong buffers

  if (threadIdx.x == 0) {
    // --- Tensor DMA descriptor (ISA 8.3/8.4): 1-D tile, 18432 x f32 ---
    unsigned lds_base = (unsigned)(uintptr_t)(void*)sc; // flat[31:0] == LDS offset
    unsigned long long ga = (unsigned long long)(uintptr_t)corners;
    const unsigned td0 = N_BOX * 8u + N_BOX; // 18432 elements (corners+area)

    v4u g0;
    g0[0] = 1u;                                  // count=1 (valid), user mode
    g0[1] = lds_base;                            // lds_addr
    g0[2] = (unsigned)ga;                        // global_addr[31:0]
    g0[3] = (unsigned)((ga >> 32) & 0x01FFFFFFu) // global_addr[56:32]
            | (2u << 30);                        // type = 2 ("image")

    v8i g1;
    g1[0] = (int)(2u << 16);                     // wg_mask=0, data_size=2 (4B)
    g1[1] = (int)((td0 & 0xFFFFu) << 16);        // tensor_dim0[15:0] @ 63:48
    g1[2] = (int)(((td0 >> 16) & 0xFFFFu)        // tensor_dim0[31:16]
                  | (1u << 16));                 // tensor_dim1 = 1
    g1[3] = (int)(td0 << 16);                    // tile_dim0 @ 127:112
    g1[4] = 1;                                   // tile_dim1 = 1
    g1[5] = (int)td0;                            // tensor_dim0_stride
    g1[6] = 0;
    g1[7] = 0;

    v4i g2 = {0, 0, 0, 0};
    v4i g3 = {0, 0, 0, 0};
    v8i g4 = {0, 0, 0, 0, 0, 0, 0, 0};

    __builtin_amdgcn_tensor_load_to_lds(g0, g1, g2, g3, g4, 0);
    __builtin_amdgcn_s_wait_tensorcnt(0);
  }
  __syncthreads();

  const float* sar = sc + N_BOX * 8;
  const int tid = threadIdx.x;
  const int j = blockIdx.x * 256 + tid; // sorted column (fixed per thread)
  const int oj = order[j];
  const float thr = *thr_p;

  // clipper = column box: 4 corners + area, kept in registers for all rows
  float2 cb[4];
#pragma unroll
  for (int k = 0; k < 4; k++)
    cb[k] = make_float2(sc[oj * 8 + 2 * k], sc[oj * 8 + 2 * k + 1]);
  const float ab = sar[oj];

  // per-thread LDS polygon buffers, stride 256 float2 (conflict-free)
  float2* bufA = &ply[0][0][tid];
  float2* bufB = &ply[1][0][tid];

  for (int r = 0; r < ROWS_PER_BLK; r++) {
    int i = blockIdx.y * ROWS_PER_BLK + r; // sorted row
    int oi = order[i];

    float iou = 0.f;
    if (j > i) {
      float aa = sar[oi];
      // subject polygon = row box corners (LDS broadcast reads)
#pragma unroll
      for (int k = 0; k < 4; k++)
        bufA[k * 256] = make_float2(sc[oi * 8 + 2 * k], sc[oi * 8 + 2 * k + 1]);

      int n = 4;
      float2* cur = bufA;
      float2* nxt = bufB;
#pragma unroll
      for (int e = 0; e < 4; e++) {
        float2 p1 = cb[e];
        float2 p2 = cb[(e + 1) & 3];
        float ex = p2.x - p1.x, ey = p2.y - p1.y;
        int m = 0;
        float2 first = cur[0];
        float2 vc = first;
        float s_c = ex * (vc.y - p1.y) - ey * (vc.x - p1.x);
        for (int k = 0; k < n; k++) {
          float2 vn = (k + 1 < n) ? cur[(k + 1) * 256] : first;
          float s_n = ex * (vn.y - p1.y) - ey * (vn.x - p1.x);
          bool in_c = s_c >= 0.f;
          bool in_n = s_n >= 0.f;
          if (in_c) { nxt[m * 256] = vc; m++; }
          if (in_c != in_n) {
            float d = s_c - s_n;
            float t = s_c / ((d == 0.f) ? 1.f : d);
            nxt[m * 256] = make_float2(vc.x + t * (vn.x - vc.x),
                                       vc.y + t * (vn.y - vc.y));
            m++;
          }
          vc = vn;
          s_c = s_n;
        }
        n = m;
        float2* tmp = cur; cur = nxt; nxt = tmp;
        if (n == 0) break;
      }

      float cross = 0.f;
      if (n > 0) {
        float2 v0 = cur[0];
        float2 vc = v0;
        for (int k = 0; k < n; k++) {
          float2 vn = (k + 1 < n) ? cur[(k + 1) * 256] : v0;
          cross += vc.x * vn.y - vc.y * vn.x;
          vc = vn;
        }
      }
      float inter = 0.5f * fabsf(cross);
      iou = inter / (aa + ab - inter + 1e-6f);
    }

    bool bit = (j > i) && (iou > thr);
    unsigned bal = __builtin_amdgcn_ballot_w32(bit);
    if ((tid & 31) == 0)
      mask[i * MASK_WORDS + (j >> 5)] = bal;
  }
}

// ---------------------------------------------------------------------------
// Kernel 4: sequential greedy scan over the bitmask, LDS-resident.
// ---------------------------------------------------------------------------
__global__ void greedy_kernel(const float* __restrict__ boxes,
                              const int* __restrict__ order,
                              const unsigned* __restrict__ mask,
                              float* __restrict__ out) {
  __shared__ unsigned sup[MASK_WORDS];
  int tid = threadIdx.x; // 64 threads
  sup[tid] = 0u;
  __syncthreads();
  for (int i = 0; i < N_BOX; i++) {
    bool sup_i = (sup[i >> 5] >> (i & 31)) & 1u;
    __syncthreads();
    if (!sup_i) sup[tid] |= mask[i * MASK_WORDS + tid];
    __syncthreads();
  }
  for (int i = tid; i < N_BOX; i += 64) {
    int oi = order[i];
    bool keep = !((sup[i >> 5] >> (i & 31)) & 1u);
    out[oi] = boxes[oi * 10 + 9] * (keep ? 1.0f : 0.0f);
  }
}

// ---------------------------------------------------------------------------
extern "C" void kernel_launch(void* const* d_in, const int* in_sizes, int n_in,
                              void* d_out, int out_size, void* d_ws, size_t ws_size,
                              hipStream_t stream) {
  (void)in_sizes; (void)n_in; (void)out_size; (void)ws_size;
  const float* boxes = (const float*)d_in[0];
  const float* thr   = (const float*)d_in[1];
  float* out = (float*)d_out;

  // workspace layout (corners and area contiguous -> one TDM tile):
  // corners(64KB) | area(8KB) | order(8KB) | mask(512KB)
  float* corners = (float*)d_ws;
  float* area = corners + N_BOX * 8;
  int* order = (int*)(area + N_BOX);
  unsigned* mask = (unsigned*)(order + N_BOX);

  prep_kernel<<<N_BOX / 256, 256, 0, stream>>>(boxes, corners, area);
  rank_kernel<<<N_BOX / 256, 256, 0, stream>>>(boxes, order);
  dim3 grid(N_BOX / 256, N_BOX / ROWS_PER_BLK); // (8, 64)
  iou_mask_kernel<<<grid, 256, 0, stream>>>(corners, order, thr, mask);
  greedy_kernel<<<1, 64, 0, stream>>>(boxes, order, mask, out);
}